// GOP_71794673320393
// MI455X (gfx1250) — compile-verified
//
#include <hip/hip_runtime.h>
#include <math.h>

// out[b,u] = tanh( max_i( exp(x[b,i]*W[i,u]) - 1 ) + bias[u] )
// exp monotonic  =>  max_i(exp(a)-1) = exp(max_i a) - 1
// => tropical (max,*) GEMM M[b,u] = max_i x[b,i]*W[i,u], then cheap epilogue.
// VALU-bound (~1G f32 mul/max vs ~9MB traffic). WMMA inapplicable: matrix
// cores reduce with +, not max. CDNA5 paths: GLOBAL_LOAD_ASYNC_TO_LDS_B32
// double-buffered staging (ASYNCcnt + s_wait_asynccnt), wave32 blocks.
// This rev: launch_bounds(256,2) (round 3's min-4-waves cap caused scratch
// spills) + 32-bit offset math so staging addresses stay 1 VGPR (saddr+off)
// instead of 64-bit vaddr pairs -> lower register pressure, no spills.

constexpr int BATCH  = 2048;
constexpr int IN_DIM = 512;
constexpr int UNITS  = 512;

constexpr int B_TILE   = 64;
constexpr int U_TILE   = 64;
constexpr int I_TILE   = 32;
constexpr int NTHREADS = 256;              // 8 waves (wave32)
constexpr int N_ITILES = IN_DIM / I_TILE;  // 16

static_assert(BATCH % B_TILE == 0 && UNITS % U_TILE == 0 && IN_DIM % I_TILE == 0, "tiling");

#if __has_builtin(__builtin_amdgcn_global_load_async_to_lds_b32)
#define USE_ASYNC_BUILTIN 1
#define USE_ASYNC 1
#elif defined(__AMDGCN__)
#define USE_ASYNC_ASM 1
#define USE_ASYNC 1
#endif

typedef __attribute__((address_space(1))) int glob_i32;
typedef __attribute__((address_space(3))) int lds_i32;

__device__ __forceinline__ void stage_b32(const float* __restrict__ g, float* l) {
#if defined(USE_ASYNC_BUILTIN)
  __builtin_amdgcn_global_load_async_to_lds_b32(
      (glob_i32*)(void*)g,
      (lds_i32*)(void*)l,
      0, 0);
#elif defined(USE_ASYNC_ASM)
  unsigned lds_off = (unsigned)(size_t)(lds_i32*)(void*)l;
  unsigned long long ga = (unsigned long long)(size_t)g;
  asm volatile("global_load_async_to_lds_b32 %0, %1, off"
               :: "v"(lds_off), "v"(ga) : "memory");
#else
  *l = *g;
#endif
}

__device__ __forceinline__ void wait_async_all() {
#if defined(USE_ASYNC)
#if __has_builtin(__builtin_amdgcn_s_wait_asynccnt)
  __builtin_amdgcn_s_wait_asynccnt(0);
#else
  asm volatile("s_wait_asynccnt 0" ::: "memory");
#endif
#endif
}

__global__ __launch_bounds__(NTHREADS, 2)   // ~256-VGPR cap: no MSB spam, no spills
void expmax_tanh_kernel(const float* __restrict__ x,
                        const float* __restrict__ W,
                        const float* __restrict__ bias,
                        float* __restrict__ out) {
  // xs stored transposed [k][b]; ws natural [k][u]. Inner-loop reads are
  // 16B ds_load_b128 with a 16-lane stride-4 pattern (conflict-free,
  // broadcast across the other 16 lanes).
  __shared__ float xs[2][I_TILE][B_TILE];
  __shared__ float ws[2][I_TILE][U_TILE];

  const unsigned tid = threadIdx.x;
  const unsigned tu  = tid & 15;    // 16 threads across U
  const unsigned tb  = tid >> 4;    // 16 threads across B
  const unsigned b0  = blockIdx.y * B_TILE;
  const unsigned u0  = blockIdx.x * U_TILE;

  // Loader mapping (coalesced global reads). All offsets are 32-bit.
  const unsigned xk = tid & 31;     // x tile: lanes sweep k
  const unsigned xb = tid >> 5;     // 8 base rows, +8*j
  const unsigned wu = tid & 63;     // W tile: lanes sweep u
  const unsigned wk = tid >> 6;     // 4 base rows, +4*j

  // Per-thread base element offsets (j and tile terms are compile-time adds).
  const unsigned xbase = (b0 + xb) * IN_DIM + xk;   // + j*8*IN_DIM + t*I_TILE
  const unsigned wbase = wk * UNITS + (u0 + wu);    // + j*4*UNITS  + t*I_TILE*UNITS

  auto stage_tile = [&](unsigned t, unsigned buf) {
    const unsigned xt = xbase + t * I_TILE;
    const unsigned wt = wbase + t * (I_TILE * UNITS);
#pragma unroll
    for (unsigned j = 0; j < 8; ++j)
      stage_b32(x + (xt + j * (8u * IN_DIM)), &xs[buf][xk][xb + 8 * j]);
#pragma unroll
    for (unsigned j = 0; j < 8; ++j)
      stage_b32(W + (wt + j * (4u * UNITS)), &ws[buf][wk + 4 * j][wu]);
  };

  float acc[4][4];
#pragma unroll
  for (int r = 0; r < 4; ++r)
#pragma unroll
    for (int c = 0; c < 4; ++c) acc[r][c] = -__builtin_inff();

  stage_tile(0, 0);
  wait_async_all();
  __syncthreads();

  for (unsigned t = 0; t < N_ITILES; ++t) {
    const unsigned buf = t & 1;
    if (t + 1 < N_ITILES) stage_tile(t + 1, buf ^ 1);  // overlap with compute

    // k in pairs: acc = max3(acc, p0, p1) -> 1.5 VALU ops per element.
#pragma unroll 4
    for (int kk = 0; kk < I_TILE; kk += 2) {
      float xv0[4], wv0[4], xv1[4], wv1[4];
#pragma unroll
      for (int r = 0; r < 4; ++r) {
        xv0[r] = xs[buf][kk + 0][tb * 4 + r];
        xv1[r] = xs[buf][kk + 1][tb * 4 + r];
      }
#pragma unroll
      for (int c = 0; c < 4; ++c) {
        wv0[c] = ws[buf][kk + 0][tu * 4 + c];
        wv1[c] = ws[buf][kk + 1][tu * 4 + c];
      }
#pragma unroll
      for (int r = 0; r < 4; ++r)
#pragma unroll
        for (int c = 0; c < 4; ++c) {
          const float p0 = xv0[r] * wv0[c];
          const float p1 = xv1[r] * wv1[c];
          acc[r][c] = fmaxf(acc[r][c], fmaxf(p0, p1));  // v_max3_num_f32
        }
    }

    wait_async_all();   // next tile resident
    __syncthreads();    // all waves done reading current buffers
  }

  // Epilogue: p = exp(M) - 1;  out = tanh(p + bias). Vectorized 16B stores.
  const unsigned ub = u0 + tu * 4;
  const float4 bv = *(const float4*)(bias + ub);
#pragma unroll
  for (int r = 0; r < 4; ++r) {
    const unsigned b = b0 + tb * 4 + r;
    float4 o;
    o.x = tanhf(__expf(acc[r][0]) - 1.0f + bv.x);
    o.y = tanhf(__expf(acc[r][1]) - 1.0f + bv.y);
    o.z = tanhf(__expf(acc[r][2]) - 1.0f + bv.z);
    o.w = tanhf(__expf(acc[r][3]) - 1.0f + bv.w);
    *(float4*)(out + (size_t)b * UNITS + ub) = o;
  }
}

extern "C" void kernel_launch(void* const* d_in, const int* in_sizes, int n_in,
                              void* d_out, int out_size, void* d_ws, size_t ws_size,
                              hipStream_t stream) {
  const float* x    = (const float*)d_in[0];
  const float* W    = (const float*)d_in[1];   // (1, IN_DIM, UNITS) => contiguous (IN_DIM, UNITS)
  const float* bias = (const float*)d_in[2];
  float* out        = (float*)d_out;

  dim3 grid(UNITS / U_TILE, BATCH / B_TILE);   // (8, 32) = 256 workgroups
  dim3 block(NTHREADS);
  hipLaunchKernelGGL(expmax_tanh_kernel, grid, block, 0, stream, x, W, bias, out);
}